// ColbertLoss_48223892799667
// MI455X (gfx1250) — compile-verified
//
#include <hip/hip_runtime.h>

typedef float v2f __attribute__((ext_vector_type(2)));
typedef float v8f __attribute__((ext_vector_type(8)));

#define NC 64      // texts
#define NT 77      // tokens per text
#define NI 64      // images
#define NP 197     // patches per image
#define ND 512     // embed dim
#define TPAD 80    // padded T (5 x 16)
#define PPAD 208   // padded P (13 x 16)
#define KC 64      // K chunk staged in LDS
#define TM_TILES 5
#define TN_TILES 13
#define NTILES (TM_TILES * TN_TILES)   // 65
#define NWAVES 8
#define MAX_TILES_PER_WAVE 9           // ceil(65/8)

// One workgroup per (c,i) pair: WMMA-f32 GEMM of text[c] (77x512) x img[i]^T
// (512x197) into an 80x208 LDS block, then masked MaxSim reductions.
__global__ __launch_bounds__(256) void colbert_scores_kernel(
    const float* __restrict__ text_emb,   // [C][T][D]
    const float* __restrict__ img_emb,    // [I][P][D]
    const float* __restrict__ scale_ptr,  // [1]
    float* __restrict__ score_p,          // [C*I]  sum_p max_t
    float* __restrict__ score_t)          // [C*I]  sum_t max_p
{
  __shared__ float lds_a[TPAD * KC];     // 20.0 KB  (text chunk, zero padded)
  __shared__ float lds_b[PPAD * KC];     // 52.0 KB  (image chunk, zero padded)
  __shared__ float lds_sim[TPAD * PPAD]; // 65.0 KB  (sim block)
  __shared__ float lds_red[256];

  const int tid  = threadIdx.x;
  const int wave = tid >> 5;
  const int lane = tid & 31;
  const int blk  = blockIdx.x;
  const int c = blk >> 6;       // I == 64
  const int i = blk & 63;

  const float* At = text_emb + (size_t)c * NT * ND;
  const float* Bi = img_emb  + (size_t)i * NP * ND;

  v8f acc[MAX_TILES_PER_WAVE];
#pragma unroll
  for (int j = 0; j < MAX_TILES_PER_WAVE; ++j) {
    v8f z = {};
    acc[j] = z;
  }

  const int lhalf = lane >> 4;  // 0 or 1 -> K offset +0 / +2 in the 16x16x4 frags
  const int lmod  = lane & 15;  // M (for A) / N (for B) index within the tile

  for (int kc = 0; kc < ND; kc += KC) {
    // Stage A chunk [TPAD][KC], zero rows t >= NT
    for (int idx = tid; idx < TPAD * KC; idx += 256) {
      int t  = idx >> 6;        // KC == 64
      int kk = idx & 63;
      lds_a[idx] = (t < NT) ? At[(size_t)t * ND + kc + kk] : 0.0f;
    }
    // Stage B chunk [PPAD][KC], zero rows p >= NP
    for (int idx = tid; idx < PPAD * KC; idx += 256) {
      int p  = idx >> 6;
      int kk = idx & 63;
      lds_b[idx] = (p < NP) ? Bi[(size_t)p * ND + kc + kk] : 0.0f;
    }
    __syncthreads();

#pragma unroll
    for (int j = 0; j < MAX_TILES_PER_WAVE; ++j) {
      int tile = wave + j * NWAVES;
      if (tile < NTILES) {
        int tm = tile / TN_TILES;
        int tn = tile % TN_TILES;
        // A frag: lane m = lmod, K pair at lhalf*2 (+0,+1); B frag symmetric in N.
        const v2f* arow = (const v2f*)&lds_a[(tm * 16 + lmod) * KC + lhalf * 2];
        const v2f* brow = (const v2f*)&lds_b[(tn * 16 + lmod) * KC + lhalf * 2];
#pragma unroll
        for (int kk = 0; kk < KC; kk += 4) {
          v2f a = arow[kk >> 1];
          v2f b = brow[kk >> 1];
          acc[j] = __builtin_amdgcn_wmma_f32_16x16x4_f32(
              false, a, false, b, (short)0, acc[j], false, false);
        }
      }
    }
    __syncthreads();
  }

  // Spill sim tiles to LDS. C/D layout: VGPR r, lanes 0-15 -> M=r, lanes 16-31 -> M=r+8.
#pragma unroll
  for (int j = 0; j < MAX_TILES_PER_WAVE; ++j) {
    int tile = wave + j * NWAVES;
    if (tile < NTILES) {
      int tm = tile / TN_TILES;
      int tn = tile % TN_TILES;
      int col = tn * 16 + lmod;
#pragma unroll
      for (int r = 0; r < 8; ++r) {
        int row = tm * 16 + r + lhalf * 8;
        lds_sim[row * PPAD + col] = acc[j][r];
      }
    }
  }
  __syncthreads();

  const float s = scale_ptr[0];

  // score_t = s * sum_t max_p  (mask: only t < NT, p < NP; padded entries never read)
  float part = 0.0f;
  if (tid < NT) {
    float m = -3.4e38f;
    const float* rowp = &lds_sim[tid * PPAD];
    for (int p = 0; p < NP; ++p) m = fmaxf(m, rowp[p]);
    part = m;
  }
  lds_red[tid] = part;
  __syncthreads();
  for (int off = 128; off > 0; off >>= 1) {
    if (tid < off) lds_red[tid] += lds_red[tid + off];
    __syncthreads();
  }
  if (tid == 0) score_t[blk] = s * lds_red[0];
  __syncthreads();

  // score_p = s * sum_p max_t
  part = 0.0f;
  if (tid < NP) {
    float m = -3.4e38f;
    for (int t = 0; t < NT; ++t) m = fmaxf(m, lds_sim[t * PPAD + tid]);
    part = m;
  }
  lds_red[tid] = part;
  __syncthreads();
  for (int off = 128; off > 0; off >>= 1) {
    if (tid < off) lds_red[tid] += lds_red[tid + off];
    __syncthreads();
  }
  if (tid == 0) score_p[blk] = s * lds_red[0];
}

// Tiny epilogue: 4 softmax/diag/log losses over the 64x64 score matrices.
__global__ __launch_bounds__(64) void colbert_loss_kernel(
    const float* __restrict__ score_p,
    const float* __restrict__ score_t,
    float* __restrict__ out)
{
  __shared__ float red[64];
  const int k = threadIdx.x;   // 0..63
  const float EPS = 1e-8f;
  float total = 0.0f;

  for (int which = 0; which < 2; ++which) {
    const float* S = which ? score_t : score_p;
    // axis 0: softmax over c at column i = k
    {
      float mx = -3.4e38f;
      for (int cc = 0; cc < 64; ++cc) mx = fmaxf(mx, S[cc * 64 + k]);
      float se = 0.0f;
      for (int cc = 0; cc < 64; ++cc) se += expf(S[cc * 64 + k] - mx);
      float sm = expf(S[k * 64 + k] - mx) / se;
      total += -logf(sm + EPS);
    }
    // axis 1: softmax over i at row c = k
    {
      float mx = -3.4e38f;
      for (int ii = 0; ii < 64; ++ii) mx = fmaxf(mx, S[k * 64 + ii]);
      float se = 0.0f;
      for (int ii = 0; ii < 64; ++ii) se += expf(S[k * 64 + ii] - mx);
      float sm = expf(S[k * 64 + k] - mx) / se;
      total += -logf(sm + EPS);
    }
  }

  red[k] = total;
  __syncthreads();
  for (int off = 32; off > 0; off >>= 1) {
    if (k < off) red[k] += red[k + off];
    __syncthreads();
  }
  if (k == 0) out[0] = red[0] * (1.0f / 256.0f);  // mean over 4 losses x 64 diag
}

extern "C" void kernel_launch(void* const* d_in, const int* in_sizes, int n_in,
                              void* d_out, int out_size, void* d_ws, size_t ws_size,
                              hipStream_t stream) {
  (void)in_sizes; (void)n_in; (void)out_size; (void)ws_size;
  // setup_inputs order: image_features[0], text_features[1] (unused),
  // image_embeddings[2], text_embeddings[3], logit_scale[4]
  const float* img_emb  = (const float*)d_in[2];
  const float* text_emb = (const float*)d_in[3];
  const float* scale    = (const float*)d_in[4];

  float* sp = (float*)d_ws;            // [64*64] score_p
  float* st = sp + NC * NI;            // [64*64] score_t

  colbert_scores_kernel<<<NC * NI, 256, 0, stream>>>(text_emb, img_emb, scale, sp, st);
  colbert_loss_kernel<<<1, 64, 0, stream>>>(sp, st, (float*)d_out);
}